// customedLoss_34565896798875
// MI455X (gfx1250) — compile-verified
//
#include <hip/hip_runtime.h>

#define H_ 512
#define W_ 512
#define TILE 32
#define SSEG_P 36   // seg tile with halo 2
#define SPR_P  34   // pred/S tile with halo 1
#define NTHREADS 256

__device__ __forceinline__ bool is_killed(int i, int j) {
    // replicates buggy eight_adjacency fall-through
    return ((i == H_ - 1) && (j > 0)) || ((j == W_ - 1) && (i > 0));
}

// CDNA5 async global->LDS copy (VGLOBAL op 96), tracked by ASYNCcnt.
// %0 = VGPR with LDS byte address, %1 = VGPR pair with 64-bit global VA.
__device__ __forceinline__ void async_load_f32(const float* lds_slot, const float* gaddr) {
    unsigned lds_byte_addr = (unsigned)(size_t)lds_slot;  // low 32 bits == LDS offset
    asm volatile("global_load_async_to_lds_b32 %0, %1, off"
                 :: "v"(lds_byte_addr), "v"(gaddr)
                 : "memory");
}

__device__ __forceinline__ void wait_async0() {
#if __has_builtin(__builtin_amdgcn_s_wait_asynccnt)
    __builtin_amdgcn_s_wait_asynccnt(0);
#else
    asm volatile("s_wait_asynccnt 0" ::: "memory");
#endif
}

__device__ __forceinline__ float nbr8(const float* t, int rr, int cc, int pitch) {
    return t[(rr - 1) * pitch + (cc - 1)] + t[(rr - 1) * pitch + cc] + t[(rr - 1) * pitch + (cc + 1)]
         + t[ rr      * pitch + (cc - 1)]                            + t[ rr      * pitch + (cc + 1)]
         + t[(rr + 1) * pitch + (cc - 1)] + t[(rr + 1) * pitch + cc] + t[(rr + 1) * pitch + (cc + 1)];
}

__global__ void loss_tile_kernel(const float* __restrict__ pred,
                                 const float* __restrict__ seg,
                                 float* __restrict__ psum,
                                 float* __restrict__ pcnt) {
    __shared__ float sseg[SSEG_P * SSEG_P];
    __shared__ float spred[SPR_P * SPR_P];
    __shared__ float sS[SPR_P * SPR_P];
    __shared__ float wsum[NTHREADS / 32];
    __shared__ float wcnt[NTHREADS / 32];

    const int tx  = threadIdx.x;                 // 0..31
    const int ty  = threadIdx.y;                 // 0..7
    const int tid = ty * TILE + tx;              // 0..255
    const int b   = blockIdx.z;
    const int gi0 = blockIdx.y * TILE;
    const int gj0 = blockIdx.x * TILE;
    const float* __restrict__ segb  = seg  + (size_t)b * H_ * W_;
    const float* __restrict__ predb = pred + (size_t)b * H_ * W_;

    // ---- Phase 0: zero halo slots, only needed for image-border blocks ----
    const bool edge = (gi0 == 0) || (gj0 == 0) || (gi0 == H_ - TILE) || (gj0 == W_ - TILE);
    if (edge) {  // block-uniform branch
        sseg[tid] = 0.f; sseg[tid + 256] = 0.f; sseg[tid + 512] = 0.f;
        sseg[tid + 768] = 0.f; sseg[tid + 1024] = 0.f;
        if (tid < SSEG_P * SSEG_P - 1280) sseg[tid + 1280] = 0.f;           // 16
        spred[tid] = 0.f; spred[tid + 256] = 0.f; spred[tid + 512] = 0.f;
        spred[tid + 768] = 0.f;
        if (tid < SPR_P * SPR_P - 1024) spred[tid + 1024] = 0.f;            // 132
        __syncthreads();
    }

    // ---- Phase 1: async-stage halo'd tiles into LDS (EXEC masks OOB lanes) ----
    // seg 36x36: cols 0..31 (lane = col); rows ty+8k (k<4) + guarded row ty+32
    {
        const int gj = gj0 - 2 + tx;
#pragma unroll
        for (int k = 0; k < 4; ++k) {
            const int r = ty + 8 * k, gi = gi0 - 2 + r;
            if ((unsigned)gi < (unsigned)H_ && (unsigned)gj < (unsigned)W_)
                async_load_f32(&sseg[r * SSEG_P + tx], segb + (size_t)gi * W_ + gj);
        }
        if (ty < 4) {
            const int r = ty + 32, gi = gi0 - 2 + r;
            if ((unsigned)gi < (unsigned)H_ && (unsigned)gj < (unsigned)W_)
                async_load_f32(&sseg[r * SSEG_P + tx], segb + (size_t)gi * W_ + gj);
        }
        // tail: cols 32..35 x 36 rows; r = tid>>2, c = 32+(tid&3)
        if (tid < 4 * SSEG_P) {
            const int r = tid >> 2, c = 32 + (tid & 3);
            const int gi = gi0 - 2 + r, gj2 = gj0 - 2 + c;
            if ((unsigned)gi < (unsigned)H_ && (unsigned)gj2 < (unsigned)W_)
                async_load_f32(&sseg[r * SSEG_P + c], segb + (size_t)gi * W_ + gj2);
        }
    }
    // pred 34x34: cols 0..31; rows ty+8k (k<4) + guarded row ty+32 (ty<2)
    {
        const int gj = gj0 - 1 + tx;
#pragma unroll
        for (int k = 0; k < 4; ++k) {
            const int r = ty + 8 * k, gi = gi0 - 1 + r;
            if ((unsigned)gi < (unsigned)H_ && (unsigned)gj < (unsigned)W_)
                async_load_f32(&spred[r * SPR_P + tx], predb + (size_t)gi * W_ + gj);
        }
        if (ty < 2) {
            const int r = ty + 32, gi = gi0 - 1 + r;
            if ((unsigned)gi < (unsigned)H_ && (unsigned)gj < (unsigned)W_)
                async_load_f32(&spred[r * SPR_P + tx], predb + (size_t)gi * W_ + gj);
        }
        // tail: cols 32..33 x 34 rows; r = tid>>1, c = 32+(tid&1)
        if (tid < 2 * SPR_P) {
            const int r = tid >> 1, c = 32 + (tid & 1);
            const int gi = gi0 - 1 + r, gj2 = gj0 - 1 + c;
            if ((unsigned)gi < (unsigned)H_ && (unsigned)gj2 < (unsigned)W_)
                async_load_f32(&spred[r * SPR_P + c], predb + (size_t)gi * W_ + gj2);
        }
    }
    wait_async0();
    __syncthreads();

    // ---- Phase 2: S = (killed) 8-neighbor sum of seg on the halo-1 region ----
    // sS (r,c) <-> global (gi0-1+r, gj0-1+c); sseg center = (r+1, c+1)
    {
#pragma unroll
        for (int k = 0; k < 4; ++k) {
            const int r = ty + 8 * k;
            const int gi = gi0 - 1 + r, gj = gj0 - 1 + tx;
            float s = 0.f;
            if ((unsigned)gi < (unsigned)H_ && (unsigned)gj < (unsigned)W_ && !is_killed(gi, gj))
                s = nbr8(sseg, r + 1, tx + 1, SSEG_P);
            sS[r * SPR_P + tx] = s;
        }
        if (ty < 2) {
            const int r = ty + 32;
            const int gi = gi0 - 1 + r, gj = gj0 - 1 + tx;
            float s = 0.f;
            if ((unsigned)gi < (unsigned)H_ && (unsigned)gj < (unsigned)W_ && !is_killed(gi, gj))
                s = nbr8(sseg, r + 1, tx + 1, SSEG_P);
            sS[r * SPR_P + tx] = s;
        }
        if (tid < 2 * SPR_P) {
            const int r = tid >> 1, c = 32 + (tid & 1);
            const int gi = gi0 - 1 + r, gj = gj0 - 1 + c;
            float s = 0.f;
            if ((unsigned)gi < (unsigned)H_ && (unsigned)gj < (unsigned)W_ && !is_killed(gi, gj))
                s = nbr8(sseg, r + 1, c + 1, SSEG_P);
            sS[r * SPR_P + c] = s;
        }
    }
    __syncthreads();

    // ---- Phase 3: per-pixel contribution; 4 pixels per thread ----
    float lsum = 0.f, lcnt = 0.f;
    const int lj = tx;
#pragma unroll
    for (int k = 0; k < 4; ++k) {
        const int li = ty + 8 * k;
        const int gi = gi0 + li, gj = gj0 + lj;
        const float pc = spred[(li + 1) * SPR_P + (lj + 1)];
        float t1 = 0.f, t2 = 0.f;
#pragma unroll
        for (int di = -1; di <= 1; ++di) {
#pragma unroll
            for (int dj = -1; dj <= 1; ++dj) {
                if (di == 0 && dj == 0) continue;
                const float tv = sseg[(li + 2 + di) * SSEG_P + (lj + 2 + dj)];
                const float pv = spred[(li + 1 + di) * SPR_P + (lj + 1 + dj)];
                const float Sn = sS  [(li + 1 + di) * SPR_P + (lj + 1 + dj)];
                const float w  = (tv != 0.f) ? Sn : 0.f;
                t1 += w * fabsf(tv - pv);
                t2 += w;
            }
        }
        const float contrib = is_killed(gi, gj) ? 0.f : t1 / (t2 + 1.f);
        if (pc >= 0.8f) { lsum += contrib; lcnt += 1.f; }
    }

    // ---- Reduction: in-wave shuffles (wave32, fixed order) + 8-entry LDS combine ----
#pragma unroll
    for (int off = 16; off > 0; off >>= 1) {
        lsum += __shfl_down(lsum, off, 32);
        lcnt += __shfl_down(lcnt, off, 32);
    }
    const int wave = tid >> 5, lane = tid & 31;
    if (lane == 0) { wsum[wave] = lsum; wcnt[wave] = lcnt; }
    __syncthreads();
    if (tid == 0) {
        float S = 0.f, C = 0.f;
#pragma unroll
        for (int w = 0; w < NTHREADS / 32; ++w) { S += wsum[w]; C += wcnt[w]; }
        const int pidx = b * (int)(gridDim.x * gridDim.y)
                       + (int)blockIdx.y * (int)gridDim.x + (int)blockIdx.x;
        psum[pidx] = S;
        pcnt[pidx] = C;
    }
}

__global__ void loss_final_kernel(const float* __restrict__ psum,
                                  const float* __restrict__ pcnt,
                                  const int* __restrict__ bn_p,
                                  float* __restrict__ out,
                                  int tilesPerImage) {
    __shared__ float rs[NTHREADS], rc[NTHREADS];
    const int tid = threadIdx.x;
    const int bn = *bn_p;
    float loss = 0.f;
    for (int b = 0; b < bn; ++b) {
        float s = 0.f, c = 0.f;
        for (int t = tid; t < tilesPerImage; t += NTHREADS) {
            s += psum[(size_t)b * tilesPerImage + t];
            c += pcnt[(size_t)b * tilesPerImage + t];
        }
        rs[tid] = s; rc[tid] = c;
        __syncthreads();
#pragma unroll
        for (int k = NTHREADS / 2; k > 0; k >>= 1) {
            if (tid < k) { rs[tid] += rs[tid + k]; rc[tid] += rc[tid + k]; }
            __syncthreads();
        }
        if (tid == 0) loss = (loss + rs[0]) / rc[0];   // running-division scan (faithful)
        __syncthreads();
    }
    if (tid == 0) out[0] = loss / (float)bn;
}

extern "C" void kernel_launch(void* const* d_in, const int* in_sizes, int n_in,
                              void* d_out, int out_size, void* d_ws, size_t ws_size,
                              hipStream_t stream) {
    const float* pred = (const float*)d_in[0];   // out_image  [B,1,H,W] f32
    const float* seg  = (const float*)d_in[1];   // segment_image
    const int*   bn   = (const int*)d_in[2];     // batch_num (device scalar)
    float* out = (float*)d_out;

    const int total = in_sizes[0];
    const int B = total / (H_ * W_);
    const int tcx = W_ / TILE, tcy = H_ / TILE;
    const int tiles = tcx * tcy;

    float* psum = (float*)d_ws;                  // B*tiles floats
    float* pcnt = psum + (size_t)B * tiles;      // B*tiles floats (32 KB total)

    dim3 grid(tcx, tcy, B), block(TILE, 8, 1);
    loss_tile_kernel<<<grid, block, 0, stream>>>(pred, seg, psum, pcnt);
    loss_final_kernel<<<1, NTHREADS, 0, stream>>>(psum, pcnt, bn, out, tiles);
}